// DecoderLSTM_35270271435537
// MI455X (gfx1250) — compile-verified
//
#include <hip/hip_runtime.h>
#include <hip/hip_bf16.h>

// ---------------------------------------------------------------------------
// DecoderLSTM beam search for MI455X (gfx1250, wave32).
// All big GEMMs (M=64) are HBM-bound -> stream weights once per step and
// compute with V_WMMA_F32_16X16X4_F32 (fp32 keeps reference numerics).
// Inner loop: dual independent WMMA accumulator chains + unconditional
// speculative NT prefetch (OOB prefetches are silently dropped per ISA).
// ---------------------------------------------------------------------------

typedef __attribute__((ext_vector_type(2))) float v2f;
typedef __attribute__((ext_vector_type(8))) float v8f;

#define KBEAM 64
#define MAXLEN 30
#define SEQW (MAXLEN + 1)          // 31
#define PDIM 196
#define ADIM 1024
#define CDIM 2048
#define DDIM 2048
#define EDIM 1024
#define VDIM 32000
#define NEGC (-1.0e9f)
#define EOS_TOK 2
#define START_TOK 1
#define PFROWS 64u                 // prefetch distance in K-rows

__device__ __forceinline__ float sigf(float v) { return 1.0f / (1.0f + __expf(-v)); }

// One 16x16 output tile per wave; dual accumulator chains over K.
// A[64,K] row-major, W[K,N] row-major, D[64,N].
__device__ __forceinline__ void wmma_tile_accum(const float* __restrict__ Ar,
                                                const float* __restrict__ W,
                                                unsigned K, unsigned N,
                                                unsigned bcol, unsigned hi,
                                                v8f& acc0, v8f& acc1) {
  const unsigned ko = 2u * hi;
  for (unsigned k = 0; k < K; k += 8) {
    v2f a0, b0, a1, b1;
    a0.x = Ar[k + ko];
    a0.y = Ar[k + ko + 1];
    b0.x = W[(k + ko) * N + bcol];
    b0.y = W[(k + ko + 1) * N + bcol];
    a1.x = Ar[k + 4 + ko];
    a1.y = Ar[k + 4 + ko + 1];
    b1.x = W[(k + 4 + ko) * N + bcol];
    b1.y = W[(k + 4 + ko + 1) * N + bcol];
    // Speculative streaming prefetch of the weight rows PFROWS ahead.
    // OOB addresses are dropped silently (speculative prefetch, ISA 10.5).
    __builtin_prefetch(&W[(k + PFROWS + ko) * N + bcol], 0, 0);
    acc0 = __builtin_amdgcn_wmma_f32_16x16x4_f32(false, a0, false, b0,
                                                 (short)0, acc0, false, false);
    acc1 = __builtin_amdgcn_wmma_f32_16x16x4_f32(false, a1, false, b1,
                                                 (short)0, acc1, false, false);
  }
}

// ---------------------------------------------------------------------------
// D[64,N] = A[64,K] @ W[K,N] + bias[N]
// 256 threads = 8 waves; wave w -> mtile = w&3, ntile = 2*blockIdx.x + (w>>2)
// ---------------------------------------------------------------------------
__global__ void wmma_gemm_bias(const float* __restrict__ A,
                               const float* __restrict__ W,
                               const float* __restrict__ bias,
                               float* __restrict__ D, unsigned K, unsigned N) {
  const unsigned wave  = threadIdx.x >> 5;
  const unsigned lane  = threadIdx.x & 31;
  const unsigned mbase = (wave & 3) * 16;
  const unsigned nbase = (blockIdx.x * 2 + (wave >> 2)) * 16;
  const unsigned l15   = lane & 15;
  const unsigned hi    = lane >> 4;          // 0 or 1
  const unsigned arow  = mbase + l15;
  const unsigned bcol  = nbase + l15;
  v8f acc0 = {}, acc1 = {};
  wmma_tile_accum(A + arow * K, W, K, N, bcol, hi, acc0, acc1);
  const float bv = bias[bcol];
#pragma unroll
  for (int v = 0; v < 8; ++v) {
    const unsigned row = mbase + v + 8 * hi;
    D[row * N + bcol] = acc0[v] + acc1[v] + bv;
  }
}

// D[64,N] = A1[64,K1]@W1 + A2[64,K2]@W2 + b1 + b2   (LSTM pre-activations)
__global__ void wmma_gemm2_bias(const float* __restrict__ A1,
                                const float* __restrict__ W1,
                                const float* __restrict__ b1,
                                const float* __restrict__ A2,
                                const float* __restrict__ W2,
                                const float* __restrict__ b2,
                                float* __restrict__ D,
                                unsigned K1, unsigned K2, unsigned N) {
  const unsigned wave  = threadIdx.x >> 5;
  const unsigned lane  = threadIdx.x & 31;
  const unsigned mbase = (wave & 3) * 16;
  const unsigned nbase = (blockIdx.x * 2 + (wave >> 2)) * 16;
  const unsigned l15   = lane & 15;
  const unsigned hi    = lane >> 4;
  const unsigned arow  = mbase + l15;
  const unsigned bcol  = nbase + l15;
  v8f acc0 = {}, acc1 = {};
  wmma_tile_accum(A1 + arow * K1, W1, K1, N, bcol, hi, acc0, acc1);
  wmma_tile_accum(A2 + arow * K2, W2, K2, N, bcol, hi, acc0, acc1);
  const float bv = b1[bcol] + b2[bcol];
#pragma unroll
  for (int v = 0; v < 8; ++v) {
    const unsigned row = mbase + v + 8 * hi;
    D[row * N + bcol] = acc0[v] + acc1[v] + bv;
  }
}

// ---------------------------------------------------------------------------
// One-time precompute: att1 = enc0 @ enc_att_w + b     [196,1024]
// ---------------------------------------------------------------------------
__global__ void k_att1(const float* __restrict__ enc0,
                       const float* __restrict__ Wa,
                       const float* __restrict__ ba,
                       float* __restrict__ att1) {
  const int idx = blockIdx.x * 256 + threadIdx.x;   // 196*1024 exact
  const int p = idx >> 10, a = idx & 1023;
  const float* e = enc0 + p * CDIM;
  float s = ba[a];
  for (int c0 = 0; c0 < CDIM; ++c0) s += e[c0] * Wa[c0 * ADIM + a];
  att1[idx] = s;
}

__global__ void k_mean(const float* __restrict__ enc0, float* __restrict__ mean) {
  const int c0 = blockIdx.x * 256 + threadIdx.x;    // 2048
  float s = 0.f;
  for (int p = 0; p < PDIM; ++p) s += enc0[p * CDIM + c0];
  mean[c0] = s * (1.0f / (float)PDIM);
}

__global__ void k_init(const float* __restrict__ mean,
                       const float* __restrict__ ihw, const float* __restrict__ ihb,
                       const float* __restrict__ icw, const float* __restrict__ icb,
                       float* __restrict__ h, float* __restrict__ c,
                       float* __restrict__ scores, int* __restrict__ prev,
                       int* __restrict__ seqsA) {
  const int d = blockIdx.x * 256 + threadIdx.x;     // 2048
  float hv = ihb[d], cv = icb[d];
  for (int cc = 0; cc < CDIM; ++cc) {
    const float m = mean[cc];
    hv += m * ihw[cc * DDIM + d];
    cv += m * icw[cc * DDIM + d];
  }
  for (int kk = 0; kk < KBEAM; ++kk) {
    h[kk * DDIM + d] = hv;
    c[kk * DDIM + d] = cv;
  }
  if (d < KBEAM) { scores[d] = (d == 0) ? 0.f : NEGC; prev[d] = START_TOK; }
  if (d < KBEAM * SEQW) seqsA[d] = ((d % SEQW) == 0) ? START_TOK : 0;
}

// ---------------------------------------------------------------------------
// Per-beam attention: e -> softmax -> awe; gate; build x = [emb, gate*awe]
// ---------------------------------------------------------------------------
__global__ void k_attend(const float* __restrict__ att1,
                         const float* __restrict__ att2,
                         const float* __restrict__ full_w,
                         const float* __restrict__ full_b,
                         const float* __restrict__ enc0,
                         const float* __restrict__ gatebuf,
                         const float* __restrict__ embedding,
                         const int* __restrict__ prev,
                         float* __restrict__ x) {
  const int k = blockIdx.x, tid = threadIdx.x;
  __shared__ float sh[256];
  float e = -3.0e38f;
  if (tid < PDIM) {
    float s = full_b[0];
    const float* a1 = att1 + tid * ADIM;
    const float* a2 = att2 + k * ADIM;
    for (int a = 0; a < ADIM; ++a) {
      float r = a1[a] + a2[a];
      r = r > 0.f ? r : 0.f;
      s += r * full_w[a];
    }
    e = s;
  }
  sh[tid] = e;
  __syncthreads();
  for (int s2 = 128; s2 > 0; s2 >>= 1) {
    if (tid < s2) sh[tid] = fmaxf(sh[tid], sh[tid + s2]);
    __syncthreads();
  }
  const float mx = sh[0];
  __syncthreads();
  const float ex = (tid < PDIM) ? __expf(e - mx) : 0.f;
  sh[tid] = ex;
  __syncthreads();
  for (int s2 = 128; s2 > 0; s2 >>= 1) {
    if (tid < s2) sh[tid] += sh[tid + s2];
    __syncthreads();
  }
  const float inv = 1.0f / sh[0];
  __syncthreads();
  sh[tid] = ex * inv;       // alpha
  __syncthreads();
  const int pv = prev[k];
  for (int ec = tid; ec < EDIM; ec += 256)
    x[k * (EDIM + CDIM) + ec] = embedding[pv * EDIM + ec];
  for (int d = tid; d < CDIM; d += 256) {
    float s = 0.f;
    for (int p = 0; p < PDIM; ++p) s += sh[p] * enc0[p * CDIM + d];
    const float g = sigf(gatebuf[k * CDIM + d]);
    x[k * (EDIM + CDIM) + EDIM + d] = g * s;
  }
}

__global__ void k_lstm(const float* __restrict__ g4, const float* __restrict__ cin,
                       float* __restrict__ h2, float* __restrict__ c2) {
  const int idx = blockIdx.x * 256 + threadIdx.x;   // 64*2048
  const int k = idx >> 11, d = idx & 2047;
  const float* g = g4 + k * 4 * DDIM;
  const float si = sigf(g[d]);
  const float sf = sigf(g[DDIM + d]);
  const float tg = tanhf(g[2 * DDIM + d]);
  const float so = sigf(g[3 * DDIM + d]);
  const float cc = sf * cin[idx] + si * tg;
  c2[idx] = cc;
  h2[idx] = so * tanhf(cc);
}

// ---------------------------------------------------------------------------
// Per-beam: log-softmax + score/frozen combine (in-place), then top-64
// Thread t owns blocked slice [t*125, t*125+125); 256*125 == 32000.
// ---------------------------------------------------------------------------
__global__ void k_topk_beam(float* __restrict__ logits,
                            const float* __restrict__ scores,
                            const int* __restrict__ prev,
                            float* __restrict__ cand_v, int* __restrict__ cand_i) {
  const int k = blockIdx.x, tid = threadIdx.x;
  float* row = logits + (long)k * VDIM;
  __shared__ float rv[256];
  __shared__ int ri[256];
  // max
  float m = -3.0e38f;
  for (int i = tid; i < VDIM; i += 256) m = fmaxf(m, row[i]);
  rv[tid] = m; __syncthreads();
  for (int s = 128; s > 0; s >>= 1) { if (tid < s) rv[tid] = fmaxf(rv[tid], rv[tid + s]); __syncthreads(); }
  const float mx = rv[0]; __syncthreads();
  // sum exp
  float sm = 0.f;
  for (int i = tid; i < VDIM; i += 256) sm += __expf(row[i] - mx);
  rv[tid] = sm; __syncthreads();
  for (int s = 128; s > 0; s >>= 1) { if (tid < s) rv[tid] += rv[tid + s]; __syncthreads(); }
  const float lse = mx + __logf(rv[0]); __syncthreads();

  const bool frozen = (prev[k] == EOS_TOK);
  const float sc = scores[k];
  const int S = VDIM / 256;                // 125
  const int base = tid * S;
  float bestv = -3.0e38f; int besti = base;
  for (int j = 0; j < S; ++j) {
    const int i = base + j;
    const float val = frozen ? (sc + ((i == EOS_TOK) ? 0.f : NEGC))
                             : (sc + row[i] - lse);
    row[i] = val;
    if (val > bestv) { bestv = val; besti = i; }
  }
  unsigned long long m0 = 0ull, m1 = 0ull;
  for (int r = 0; r < KBEAM; ++r) {
    rv[tid] = bestv; ri[tid] = tid; __syncthreads();
    for (int s = 128; s > 0; s >>= 1) {
      if (tid < s && rv[tid + s] > rv[tid]) { rv[tid] = rv[tid + s]; ri[tid] = ri[tid + s]; }
      __syncthreads();
    }
    const int wt = ri[0]; const float wv = rv[0]; __syncthreads();
    if (tid == 0) cand_v[k * KBEAM + r] = wv;
    if (tid == wt) {
      cand_i[k * KBEAM + r] = besti;
      const int j = besti - base;
      if (j < 64) m0 |= 1ull << j; else m1 |= 1ull << (j - 64);
      bestv = -3.0e38f; besti = base;
      for (int j2 = 0; j2 < S; ++j2) {
        const bool used = (j2 < 64) ? ((m0 >> j2) & 1ull) : ((m1 >> (j2 - 64)) & 1ull);
        if (used) continue;
        const float val = row[base + j2];
        if (val > bestv) { bestv = val; besti = base + j2; }
      }
    }
    __syncthreads();
  }
}

// Merge 64x64 candidates -> global top-64; update scores/prev/pb and seqs.
__global__ void k_topk_final(const float* __restrict__ cand_v,
                             const int* __restrict__ cand_i,
                             const int* __restrict__ seqs_src,
                             int* __restrict__ seqs_dst,
                             float* __restrict__ scores, int* __restrict__ prev,
                             int* __restrict__ pb_out, int t) {
  __shared__ float sv[KBEAM * KBEAM];
  __shared__ float rv[256];
  __shared__ int ri[256];
  __shared__ int s_pb[KBEAM], s_nw[KBEAM];
  __shared__ float s_ts[KBEAM];
  const int tid = threadIdx.x;
  for (int i = tid; i < KBEAM * KBEAM; i += 256) sv[i] = cand_v[i];
  __syncthreads();
  for (int r = 0; r < KBEAM; ++r) {
    float bv = -3.0e38f; int bi = tid;
    for (int i = tid; i < KBEAM * KBEAM; i += 256)
      if (sv[i] > bv) { bv = sv[i]; bi = i; }
    rv[tid] = bv; ri[tid] = bi; __syncthreads();
    for (int s = 128; s > 0; s >>= 1) {
      if (tid < s && rv[tid + s] > rv[tid]) { rv[tid] = rv[tid + s]; ri[tid] = ri[tid + s]; }
      __syncthreads();
    }
    if (tid == 0) {
      const int slot = ri[0];
      s_pb[r] = slot >> 6;
      s_nw[r] = cand_i[slot];
      s_ts[r] = rv[0];
      sv[slot] = -3.0e38f;
    }
    __syncthreads();
  }
  if (tid < KBEAM) { scores[tid] = s_ts[tid]; prev[tid] = s_nw[tid]; pb_out[tid] = s_pb[tid]; }
  __syncthreads();
  for (int i = tid; i < KBEAM * SEQW; i += 256) {
    const int kk = i / SEQW, col = i % SEQW;
    int v = seqs_src[s_pb[kk] * SEQW + col];
    if (col == t + 1) v = s_nw[kk];
    seqs_dst[i] = v;
  }
}

__global__ void k_reorder(const float* __restrict__ h2, const float* __restrict__ c2,
                          const int* __restrict__ pb,
                          float* __restrict__ h, float* __restrict__ c) {
  const int idx = blockIdx.x * 256 + threadIdx.x;   // 64*2048
  const int k = idx >> 11, d = idx & 2047;
  const int src = pb[k] * DDIM + d;
  h[idx] = h2[src];
  c[idx] = c2[src];
}

__global__ void k_writeout(const int* __restrict__ seqs,
                           const float* __restrict__ scores, void* out) {
  const int i = blockIdx.x * 256 + threadIdx.x;
  if (i < KBEAM * SEQW) ((int*)out)[i] = seqs[i];
  else if (i < KBEAM * SEQW + KBEAM) ((float*)out)[i] = scores[i - KBEAM * SEQW];
}

// ---------------------------------------------------------------------------
extern "C" void kernel_launch(void* const* d_in, const int* in_sizes, int n_in,
                              void* d_out, int out_size, void* d_ws, size_t ws_size,
                              hipStream_t stream) {
  const float* enc0      = (const float*)d_in[0];
  const float* enc_att_w = (const float*)d_in[1];
  const float* enc_att_b = (const float*)d_in[2];
  const float* dec_att_w = (const float*)d_in[3];
  const float* dec_att_b = (const float*)d_in[4];
  const float* full_w    = (const float*)d_in[5];
  const float* full_b    = (const float*)d_in[6];
  const float* embedding = (const float*)d_in[7];
  const float* w_ih      = (const float*)d_in[8];
  const float* b_ih      = (const float*)d_in[9];
  const float* w_hh      = (const float*)d_in[10];
  const float* b_hh      = (const float*)d_in[11];
  const float* init_h_w  = (const float*)d_in[12];
  const float* init_h_b  = (const float*)d_in[13];
  const float* init_c_w  = (const float*)d_in[14];
  const float* init_c_b  = (const float*)d_in[15];
  const float* f_beta_w  = (const float*)d_in[16];
  const float* f_beta_b  = (const float*)d_in[17];
  const float* fc_w      = (const float*)d_in[18];
  const float* fc_b      = (const float*)d_in[19];

  // workspace layout (floats then ints)
  float* w = (float*)d_ws;
  float* att1    = w;                 w += PDIM * ADIM;      // 200704
  float* mean    = w;                 w += CDIM;
  float* h       = w;                 w += KBEAM * DDIM;
  float* c       = w;                 w += KBEAM * DDIM;
  float* h2      = w;                 w += KBEAM * DDIM;
  float* c2      = w;                 w += KBEAM * DDIM;
  float* att2    = w;                 w += KBEAM * ADIM;
  float* gatebuf = w;                 w += KBEAM * CDIM;
  float* x       = w;                 w += KBEAM * (EDIM + CDIM);
  float* g4      = w;                 w += KBEAM * 4 * DDIM;
  float* logits  = w;                 w += (size_t)KBEAM * VDIM;
  float* scores  = w;                 w += KBEAM;
  float* cand_v  = w;                 w += KBEAM * KBEAM;
  int* ip     = (int*)w;
  int* cand_i = ip;                   ip += KBEAM * KBEAM;
  int* prev   = ip;                   ip += KBEAM;
  int* pb     = ip;                   ip += KBEAM;
  int* seqsA  = ip;                   ip += KBEAM * SEQW;
  int* seqsB  = ip;                   ip += KBEAM * SEQW;

  // one-time precompute
  k_att1<<<(PDIM * ADIM) / 256, 256, 0, stream>>>(enc0, enc_att_w, enc_att_b, att1);
  k_mean<<<CDIM / 256, 256, 0, stream>>>(enc0, mean);
  k_init<<<CDIM / 256, 256, 0, stream>>>(mean, init_h_w, init_h_b, init_c_w,
                                         init_c_b, h, c, scores, prev, seqsA);

  for (int t = 0; t < MAXLEN; ++t) {
    int* seqs_src = (t & 1) ? seqsB : seqsA;
    int* seqs_dst = (t & 1) ? seqsA : seqsB;

    wmma_gemm_bias<<<ADIM / 32, 256, 0, stream>>>(h, dec_att_w, dec_att_b,
                                                  att2, DDIM, ADIM);
    wmma_gemm_bias<<<CDIM / 32, 256, 0, stream>>>(h, f_beta_w, f_beta_b,
                                                  gatebuf, DDIM, CDIM);
    k_attend<<<KBEAM, 256, 0, stream>>>(att1, att2, full_w, full_b, enc0,
                                        gatebuf, embedding, prev, x);
    wmma_gemm2_bias<<<(4 * DDIM) / 32, 256, 0, stream>>>(
        x, w_ih, b_ih, h, w_hh, b_hh, g4, EDIM + CDIM, DDIM, 4 * DDIM);
    k_lstm<<<(KBEAM * DDIM) / 256, 256, 0, stream>>>(g4, c, h2, c2);
    wmma_gemm_bias<<<VDIM / 32, 256, 0, stream>>>(h2, fc_w, fc_b,
                                                  logits, DDIM, VDIM);
    k_topk_beam<<<KBEAM, 256, 0, stream>>>(logits, scores, prev, cand_v, cand_i);
    k_topk_final<<<1, 256, 0, stream>>>(cand_v, cand_i, seqs_src, seqs_dst,
                                        scores, prev, pb, t);
    k_reorder<<<(KBEAM * DDIM) / 256, 256, 0, stream>>>(h2, c2, pb, h, c);
  }
  // after t=29 (odd), final seqs live in seqsA
  k_writeout<<<(KBEAM * SEQW + KBEAM + 255) / 256, 256, 0, stream>>>(seqsA, scores, d_out);
}